// RMSPELoss_14499809591582
// MI455X (gfx1250) — compile-verified
//
#include <hip/hip_runtime.h>
#include <math.h>

typedef __attribute__((ext_vector_type(2))) float v2f;
typedef __attribute__((ext_vector_type(8))) float v8f;
typedef __attribute__((__vector_size__(16))) int v4i;   // native vector-of-4-int

#define NUM_CLASSES 181
#define BATCH_N     131072
#define ROWS_PER_BLOCK 256
#define THREADS     256
#define CHUNK_ROWS  32
#define N_CHUNKS    (ROWS_PER_BLOCK / CHUNK_ROWS)           // 8
#define CHUNK_DW    (CHUNK_ROWS * NUM_CLASSES)              // 5792 dwords
#define CHUNK_V4    (CHUNK_DW / 4)                          // 1448 vec4 (exact: 5792 % 4 == 0)

#if __has_builtin(__builtin_amdgcn_global_load_async_to_lds_b128)
#define HAVE_ASYNC_LDS 1
typedef __attribute__((address_space(1))) v4i glb_v4i;      // global (device) AS pointer target
typedef __attribute__((address_space(3))) v4i lds_v4i;      // LDS AS pointer target
#endif

// permutation p packed as 2-bit fields: perm[p][i] = (PCODE[p] >> (2*i)) & 3
#define PC(a,b,c,d) ((unsigned)((a) | ((b) << 2) | ((c) << 4) | ((d) << 6)))
__device__ const unsigned PCODE[24] = {
    PC(0,1,2,3), PC(0,1,3,2), PC(0,2,1,3), PC(0,2,3,1), PC(0,3,1,2), PC(0,3,2,1),
    PC(1,0,2,3), PC(1,0,3,2), PC(1,2,0,3), PC(1,2,3,0), PC(1,3,0,2), PC(1,3,2,0),
    PC(2,0,1,3), PC(2,0,3,1), PC(2,1,0,3), PC(2,1,3,0), PC(2,3,0,1), PC(2,3,1,0),
    PC(3,0,1,2), PC(3,0,2,1), PC(3,1,0,2), PC(3,1,2,0), PC(3,2,0,1), PC(3,2,1,0)
};

__device__ __forceinline__ float sq_mod_diff(float a, float b) {
    const float TWO_PI = 6.283185307179586f;
    const float PI_F   = 3.14159265358979324f;
    float x = a - b + PI_F;
    float m = fmodf(x, TWO_PI);          // sign of dividend
    m += (m < 0.0f) ? TWO_PI : 0.0f;     // -> jnp.mod semantics (non-negative)
    m -= PI_F;
    return m * m;
}

__global__ __launch_bounds__(THREADS)
void rmspe_main(const float* __restrict__ logits,
                const int*   __restrict__ labels,
                float*       __restrict__ block_part)
{
    __shared__ __align__(16) int labelBuf[CHUNK_DW];       // 23168 B staging buffer
    __shared__ float costLds[ROWS_PER_BLOCK * 17];         // stride 17 -> conflict-free
    __shared__ int   idxLds[CHUNK_ROWS * 4];
    __shared__ int   cntLds[CHUNK_ROWS];
    __shared__ float wavePart[THREADS / 32];

    const int tid  = threadIdx.x;
    const int lane = tid & 31;
    const int wave = tid >> 5;
    const int half = lane >> 4;     // 0: lanes 0-15, 1: lanes 16-31
    const int l15  = lane & 15;

    const long long row = (long long)blockIdx.x * ROWS_PER_BLOCK + tid;

    // logits: only the first 4 columns of this thread's row
    const float* lg = logits + row * NUM_CLASSES;
    float da[4];
    da[0] = lg[0]; da[1] = lg[1]; da[2] = lg[2]; da[3] = lg[3];

    // ---------- Phase 1: chunked coalesced staging + parallel label scan ----------
    const float DEG2RAD = 0.017453292519943295f;
    float an[4] = {0.0f, 0.0f, 0.0f, 0.0f};

    const int* blkLabels = labels + (long long)blockIdx.x * ROWS_PER_BLOCK * NUM_CLASSES;

    for (int c = 0; c < N_CHUNKS; ++c) {
        // stage 32 rows of labels into LDS, fully coalesced 128-bit traffic
        const int* gsrc = blkLabels + c * CHUNK_DW;        // 16B aligned: CHUNK_DW % 4 == 0
        for (int j = tid; j < CHUNK_V4; j += THREADS) {
#ifdef HAVE_ASYNC_LDS
            __builtin_amdgcn_global_load_async_to_lds_b128(
                (glb_v4i*)(uintptr_t)(gsrc + 4 * j),
                (lds_v4i*)(uintptr_t)(&labelBuf[4 * j]),
                0, 0);
#else
            *(v4i*)&labelBuf[4 * j] = *(const v4i*)(gsrc + 4 * j);
#endif
        }
        if (tid < CHUNK_ROWS) cntLds[tid] = 0;
#ifdef HAVE_ASYNC_LDS
#if __has_builtin(__builtin_amdgcn_s_wait_asynccnt)
        __builtin_amdgcn_s_wait_asynccnt(0);
#else
        asm volatile("s_wait_asynccnt 0" ::: "memory");
#endif
#endif
        __syncthreads();

        // parallel scan: 8 threads per row, 23 columns each (order-free collection:
        // min over permutations is symmetric in the label-angle order)
        {
            const int rloc = tid >> 3;
            const int seg  = tid & 7;
            const int beg  = seg * 23;
            const int end  = (beg + 23 < NUM_CLASSES) ? (beg + 23) : NUM_CLASSES;
            const int* rb = &labelBuf[rloc * NUM_CLASSES];
            for (int col = beg; col < end; ++col) {
                if (rb[col] != 0) {
                    int pos = atomicAdd(&cntLds[rloc], 1);   // ds_add_rtn_u32
                    if (pos < 4) idxLds[rloc * 4 + pos] = col;
                }
            }
        }
        __syncthreads();

        // the owning thread of each row in this chunk grabs its 4 angles
        if ((tid >> 5) == c) {
            const int rloc = tid & 31;
#pragma unroll
            for (int q = 0; q < 4; ++q)
                an[q] = ((float)idxLds[rloc * 4 + q] - 90.0f) * DEG2RAD;
        }
        __syncthreads();   // protect idxLds/labelBuf before next chunk reuses them
    }

    // 4x4 cost matrix, flattened k = i*4 + j, into LDS
    float* myCost = &costLds[tid * 17];
#pragma unroll
    for (int i = 0; i < 4; ++i)
#pragma unroll
        for (int j = 0; j < 4; ++j)
            myCost[i * 4 + j] = sq_mod_diff(da[i], an[j]);

    __syncthreads();

    // ---------- Phase 2: totals = cost(16x16-flat) x P(16x24) via f32 WMMA ----------
    // B fragments (16x16x4 layout): lanes 0-15 hold K = 4s+{0,1}, lanes 16-31 K = 4s+{2,3};
    // N = lane&15 (+16 for tile 1). B[k][p] = (perm[p][k>>2] == (k&3)).
    v2f bf[2][4];
#pragma unroll
    for (int t = 0; t < 2; ++t) {
        int p = l15 + 16 * t;
        bool valid = (p < 24);
        unsigned code = PCODE[valid ? p : 0];
#pragma unroll
        for (int s = 0; s < 4; ++s) {
            unsigned pi = (code >> (2 * s)) & 3u;   // perm[p][i], i == s
            bf[t][s].x = (valid && pi == (unsigned)(2 * half    )) ? 1.0f : 0.0f;
            bf[t][s].y = (valid && pi == (unsigned)(2 * half + 1)) ? 1.0f : 0.0f;
        }
    }

    float partial = 0.0f;
#pragma unroll
    for (int rt = 0; rt < 2; ++rt) {
        const int srcRow = wave * 32 + rt * 16 + l15;   // A-row this lane supplies
        v8f c0 = {}; // perms 0..15
        v8f c1 = {}; // perms 16..23 (+ zero padding 24..31)
#pragma unroll
        for (int s = 0; s < 4; ++s) {
            // A 16x4 f32 layout: lanes 0-15 -> K = 4s+{0,1}; lanes 16-31 -> K = 4s+{2,3}
            v2f a;
            int k = 4 * s + 2 * half;
            a.x = costLds[srcRow * 17 + k];
            a.y = costLds[srcRow * 17 + k + 1];
            c0 = __builtin_amdgcn_wmma_f32_16x16x4_f32(false, a, false, bf[0][s],
                                                       (short)0, c0, false, false);
            c1 = __builtin_amdgcn_wmma_f32_16x16x4_f32(false, a, false, bf[1][s],
                                                       (short)0, c1, false, false);
        }
        // C layout: VGPR r, half h -> row M = r + 8h, col N = lane&15.
#pragma unroll
        for (int r = 0; r < 8; ++r) {
            float v = c0[r];
            float w = (l15 < 8) ? c1[r] : 3.4028235e38f;   // mask padded perms 24..31
            v = fminf(v, w);
            v = fminf(v, __shfl_xor(v, 1));
            v = fminf(v, __shfl_xor(v, 2));
            v = fminf(v, __shfl_xor(v, 4));
            v = fminf(v, __shfl_xor(v, 8));                // min within each 16-lane half
            float rmse = sqrtf(v * 0.25f);
            if (l15 == 0) partial += rmse;                 // lane 0 -> row r, lane 16 -> row r+8
        }
    }

    // wave sum (only lanes 0 and 16 carry data; fixed xor tree -> deterministic)
    partial += __shfl_xor(partial, 16);
    partial += __shfl_xor(partial, 8);
    partial += __shfl_xor(partial, 4);
    partial += __shfl_xor(partial, 2);
    partial += __shfl_xor(partial, 1);
    if (lane == 0) wavePart[wave] = partial;
    __syncthreads();

    if (tid == 0) {
        float s = 0.0f;
#pragma unroll
        for (int w = 0; w < THREADS / 32; ++w) s += wavePart[w];
        block_part[blockIdx.x] = s;
    }
}

__global__ __launch_bounds__(256)
void rmspe_reduce(const float* __restrict__ block_part, float* __restrict__ out, int n)
{
    __shared__ float sh[256];
    float s = 0.0f;
    for (int i = threadIdx.x; i < n; i += 256) s += block_part[i];
    sh[threadIdx.x] = s;
    __syncthreads();
    for (int off = 128; off > 0; off >>= 1) {
        if (threadIdx.x < off) sh[threadIdx.x] += sh[threadIdx.x + off];
        __syncthreads();
    }
    if (threadIdx.x == 0) out[0] = sh[0] * (1.0f / (float)BATCH_N);
}

extern "C" void kernel_launch(void* const* d_in, const int* in_sizes, int n_in,
                              void* d_out, int out_size, void* d_ws, size_t ws_size,
                              hipStream_t stream) {
    const float* logits = (const float*)d_in[0];
    const int*   labels = (const int*)d_in[1];
    float* partials = (float*)d_ws;                  // 512 floats of scratch
    const int nblocks = BATCH_N / ROWS_PER_BLOCK;    // 512, exact
    rmspe_main<<<nblocks, THREADS, 0, stream>>>(logits, labels, partials);
    rmspe_reduce<<<1, 256, 0, stream>>>(partials, (float*)d_out, nblocks);
}